// Decoder_22050362097675
// MI455X (gfx1250) — compile-verified
//
#include <hip/hip_runtime.h>
#include <stdint.h>

#ifndef __has_builtin
#define __has_builtin(x) 0
#endif

// Problem constants (match reference: B=256, L=1024, T=128)
constexpr int Ln = 1024;
constexpr int Tn = 128;
constexpr int TP = 132; // padded LDS row stride for trans^T: 132*4=528 B (16B aligned),
                        // lane j starts at bank 4j mod 64 -> optimal 2-phase b128 access

// LDS budget per workgroup (CDNA5 WGP has 320 KB):
//   transT : Tn*TP*4      = 67584 B
//   state  : 2*Tn*4       =  1024 B   (ping-pong Viterbi state)
//   bp     : Ln*Tn        = 131072 B  (uint8 backpointers, whole sequence resident)
//   outbuf : Ln*4         =  4096 B
//   reduce : 2*Tn*4       =  1024 B
constexpr size_t SMEM_BYTES =
    (size_t)Tn * TP * 4 + 2 * Tn * 4 + (size_t)Ln * Tn + (size_t)Ln * 4 + 2 * Tn * 4;

__global__ __launch_bounds__(Tn)
void viterbi_crf_kernel(const float* __restrict__ x,      // [B, L, T]
                        const int* __restrict__ lengths,  // [B]
                        const float* __restrict__ trans,  // [T, T]
                        int* __restrict__ out)            // [B, L]
{
    extern __shared__ char smem[];
    float*         s_transT = (float*)smem;                         // [Tn][TP] trans^T
    float*         s_state  = s_transT + Tn * TP;                   // [2][Tn]
    unsigned char* s_bp     = (unsigned char*)(s_state + 2 * Tn);   // [Ln][Tn]
    int*           s_out    = (int*)(s_bp + (size_t)Ln * Tn);       // [Ln]
    float*         s_rv     = (float*)(s_out + Ln);                 // [Tn]
    int*           s_ri     = (int*)(s_rv + Tn);                    // [Tn]

    const int j   = threadIdx.x;          // this thread owns destination state j
    const int b   = blockIdx.x;
    const int len = lengths[b];
    const size_t base = (size_t)b * Ln * Tn;

    // ---- Fill LDS with trans^T (transposed via per-lane LDS addressing) ----
#if __has_builtin(__builtin_amdgcn_global_load_async_to_lds_b32)
    {
        typedef __attribute__((address_space(1))) int* gint_t;  // global (AS1)
        typedef __attribute__((address_space(3))) int* lint_t;  // LDS    (AS3)
        for (int r = 0; r < Tn; ++r) {
            // lane j copies trans[r][j] from global memory to transT[j][r] in LDS (global reads coalesced)
            __builtin_amdgcn_global_load_async_to_lds_b32(
                (gint_t)(trans + r * Tn + j),
                (lint_t)(s_transT + j * TP + r),
                0, 0);
        }
#if __has_builtin(__builtin_amdgcn_s_wait_asynccnt)
        __builtin_amdgcn_s_wait_asynccnt(0);
#else
        asm volatile("s_wait_asynccnt 0" ::: "memory");
#endif
    }
#else
    for (int r = 0; r < Tn; ++r)
        s_transT[j * TP + r] = trans[r * Tn + j];
#endif

    // ---- t = 0: state = x[b,0,:] ----
    s_state[j] = x[base + j];
    int p = 0;
    __syncthreads();  // transT + initial state visible to all 4 waves

    // ---- Forward Viterbi scan (sequential over t, parallel over states) ----
    float xnext = (len > 1) ? x[base + Tn + j] : 0.0f;
    for (int t = 1; t < len; ++t) {
        float xt = xnext;
        if (t + 1 < len) xnext = x[base + (size_t)(t + 1) * Tn + j];
        if (t + 8 < len) __builtin_prefetch(&x[base + (size_t)(t + 8) * Tn + j], 0, 0);

        const float4* tr4 = (const float4*)(s_transT + j * TP);
        const float4* st4 = (const float4*)(s_state + p * Tn);

        // 4 independent (max, argmax) accumulators to break the VALU dep chain
        float v0 = -3.402823466e38f, v1 = v0, v2 = v0, v3 = v0;
        int g0 = 0, g1 = 0, g2 = 0, g3 = 0;
#pragma unroll
        for (int g = 0; g < Tn / 4; ++g) {
            float4 tr = tr4[g];   // ds_load_b128, 2-phase conflict-free
            float4 st = st4[g];   // ds_load_b128 broadcast (same addr all lanes)
            float c0 = st.x + tr.x;
            float c1 = st.y + tr.y;
            float c2 = st.z + tr.z;
            float c3 = st.w + tr.w;
            if (c0 > v0) { v0 = c0; g0 = g; }
            if (c1 > v1) { v1 = c1; g1 = g; }
            if (c2 > v2) { v2 = c2; g2 = g; }
            if (c3 > v3) { v3 = c3; g3 = g; }
        }
        // Merge with exact first-occurrence tie-break (matches jnp.argmax)
        int i0 = 4 * g0, i1 = 4 * g1 + 1, i2 = 4 * g2 + 2, i3 = 4 * g3 + 3;
        if (v1 > v0 || (v1 == v0 && i1 < i0)) { v0 = v1; i0 = i1; }
        if (v3 > v2 || (v3 == v2 && i3 < i2)) { v2 = v3; i2 = i3; }
        if (v2 > v0 || (v2 == v0 && i2 < i0)) { v0 = v2; i0 = i2; }

        s_bp[(size_t)t * Tn + j]   = (unsigned char)i0;
        s_state[(p ^ 1) * Tn + j]  = v0 + xt;
        __syncthreads();          // one barrier per timestep (ping-pong buffers)
        p ^= 1;
    }

    // ---- argmax over final state (first-occurrence tie-break) ----
    s_rv[j] = s_state[p * Tn + j];
    s_ri[j] = j;
    __syncthreads();
    for (int s = Tn / 2; s > 0; s >>= 1) {
        if (j < s) {
            float va = s_rv[j], vb = s_rv[j + s];
            int   ia = s_ri[j], ib = s_ri[j + s];
            if (vb > va || (vb == va && ib < ia)) { s_rv[j] = vb; s_ri[j] = ib; }
        }
        __syncthreads();
    }

    // ---- Backward pointer-chase, entirely in LDS ----
    if (j == 0) {
        int carry = s_ri[0];
        for (int t = Ln - 1; t >= 0; --t) {
            s_out[t] = (t < len) ? carry : 0;
            if (t > 0 && t < len) carry = (int)s_bp[(size_t)t * Tn + carry];
        }
    }
    __syncthreads();

    // ---- Coalesced output write ----
    for (int t = j; t < Ln; t += Tn)
        out[(size_t)b * Ln + t] = s_out[t];
}

extern "C" void kernel_launch(void* const* d_in, const int* in_sizes, int n_in,
                              void* d_out, int out_size, void* d_ws, size_t ws_size,
                              hipStream_t stream) {
    const float* x       = (const float*)d_in[0];  // inputs  [B,L,T] f32
    const int*   lengths = (const int*)d_in[1];    // lengths [B]     i32
    // d_in[2] = tags (unused by the reference's output)
    const float* trans   = (const float*)d_in[3];  // transition_params [T,T] f32
    int*         out     = (int*)d_out;            // decoded tags [B,L] i32

    const int B = in_sizes[1];                     // one workgroup per batch element

    // Allow >64KB dynamic LDS (no-op / ignored if unnecessary on this runtime)
    (void)hipFuncSetAttribute((const void*)viterbi_crf_kernel,
                              hipFuncAttributeMaxDynamicSharedMemorySize,
                              (int)SMEM_BYTES);

    viterbi_crf_kernel<<<dim3(B), dim3(Tn), SMEM_BYTES, stream>>>(x, lengths, trans, out);
}